// TextInput_7009386627295
// MI455X (gfx1250) — compile-verified
//
#include <hip/hip_runtime.h>

typedef __attribute__((ext_vector_type(2))) float v2f;
typedef __attribute__((ext_vector_type(8))) float v8f;

#define T_CTX    2048
#define DM       1024
#define BATCHSZ  4
#define NBLK     16
#define ROWS     64          // output rows per workgroup
#define NTHREADS 256         // 8 waves (wave32)

// LDS layout (floats): hT [1024][8] (cols 4..7 = 0) @ 0 | red [4][256] @ 32 KB.
// W lives entirely in VGPRs (time-invariant A operand), not in LDS.
#define LDS_HT_ELEMS  (DM * 8)
#define LDS_RED_ELEMS (4 * 256)
#define LDS_TOTAL_B   ((LDS_HT_ELEMS + LDS_RED_ELEMS) * 4)

extern "C" __global__ __launch_bounds__(NTHREADS, 1)
void recur_wmma_f32_kernel(const int* __restrict__ ids,     // [4][2048]
                           const float* __restrict__ W,     // [1024][1024]
                           const float* __restrict__ emb,   // [50257][1024]
                           float* __restrict__ out,         // [4][2048][1024]
                           unsigned* __restrict__ bar_cnt,  // ws: barrier counter (memset 0)
                           float* __restrict__ hbuf0,       // ws: h ping (memset 0 = h[-1])
                           float* __restrict__ hbuf1)       // ws: h pong
{
    extern __shared__ float lds[];
    float* hT  = lds;                    // 1024*8 (transposed h, zero-padded cols)
    float* red = lds + LDS_HT_ELEMS;     // 4*256

    const int tid  = threadIdx.x;
    const int j0   = blockIdx.x * ROWS;
    const int wave = tid >> 5;
    const int lane = tid & 31;
    const int mt   = wave & 3;               // M tile 0..3 (16 rows each)
    const int kh   = wave >> 2;              // K half 0..1 (512 each)
    const int r    = lane & 15;              // M row (A) / N col (B,C)
    const int hi   = lane >> 4;              // lane-half selects K pair / M+8

    // ---- zero-pad hT columns 4..7 (read for N>=4, never written again) ----
    for (int k = tid; k < DM; k += NTHREADS) {
        hT[k * 8 + 4] = 0.f; hT[k * 8 + 5] = 0.f;
        hT[k * 8 + 6] = 0.f; hT[k * 8 + 7] = 0.f;
    }

    const int col    = (r < BATCHSZ) ? r : 4;        // zero-pad column for N>=4
    const int kbase0 = kh * 512;

    // ---- one-time: load this wave's whole A strip of W into registers.
    //      areg[it] holds W[j0+mt*16+r][kbase0 + it*4 + 2*hi + {0,1}] — the exact
    //      per-lane v2f the WMMA A layout needs.  256 VGPRs/lane, resident for
    //      all 2048 steps (CDNA5 wave32 supports up to 1024 VGPRs/wave).
    v2f areg[128];
    {
        const float* wrow = W + (long)(j0 + mt * 16 + r) * DM + kbase0 + 2 * hi;
#pragma unroll
        for (int it = 0; it < 128; ++it) {
            areg[it].x = wrow[it * 4];
            areg[it].y = wrow[it * 4 + 1];
        }
    }
    __syncthreads();

    for (int t = 0; t < T_CTX; ++t) {
        const float* hin  = (t & 1) ? hbuf1 : hbuf0;
        float*       hout = (t & 1) ? hbuf0 : hbuf1;

        // ---- stage h[t-1] (4x1024) into LDS transposed: hT[k][n] = h[n][k] ----
        for (int i = tid; i < BATCHSZ * DM; i += NTHREADS) {
            int n = i >> 10, k = i & (DM - 1);
            hT[k * 8 + n] = hin[i];
        }
        __syncthreads();

        // ---- Y = W_tile @ h^T: 128 fp32 WMMAs, A from registers, B from LDS.
        //      Two interleaved accumulator chains hide dependent-WMMA latency.
        v8f c0 = {};
        v8f c1 = {};
#pragma unroll
        for (int it = 0; it < 64; ++it) {
            int kkA = kbase0 + it * 8 + 2 * hi;      // chain 0
            int kkB = kkA + 4;                       // chain 1
            v2f b0, b1;
            b0.x = hT[kkA * 8 + col];
            b0.y = hT[kkA * 8 + 8 + col];
            b1.x = hT[kkB * 8 + col];
            b1.y = hT[kkB * 8 + 8 + col];
            c0 = __builtin_amdgcn_wmma_f32_16x16x4_f32(
                     false, areg[2 * it], false, b0, (short)0, c0, false, false);
            c1 = __builtin_amdgcn_wmma_f32_16x16x4_f32(
                     false, areg[2 * it + 1], false, b1, (short)0, c1, false, false);
        }
        v8f c;
#pragma unroll
        for (int v = 0; v < 8; ++v) c[v] = c0[v] + c1[v];

        // ---- merge the two K-halves through LDS ----
        if (kh == 1) {
#pragma unroll
            for (int v = 0; v < 8; ++v) red[mt * 256 + v * 32 + lane] = c[v];
        }
        __syncthreads();
        if (kh == 0) {
#pragma unroll
            for (int v = 0; v < 8; ++v) c[v] += red[mt * 256 + v * 32 + lane];
            if (r < BATCHSZ) {
                int id = ids[r * T_CTX + t];
                const float* erow = emb + (long)id * DM;
#pragma unroll
                for (int v = 0; v < 8; ++v) {
                    int j = j0 + mt * 16 + v + 8 * hi;     // C row M = v + 8*hi
                    float val = c[v] + erow[j];
                    hout[r * DM + j] = val;                           // next-step state
                    out[((long)r * T_CTX + t) * DM + j] = val;        // result
                }
            }
        }

        // ---- device-wide barrier (release h writes, acquire others') ----
        __threadfence();          // each wave: stores visible at device scope
        __syncthreads();
        if (tid == 0) {
            __hip_atomic_fetch_add(bar_cnt, 1u, __ATOMIC_RELEASE,
                                   __HIP_MEMORY_SCOPE_AGENT);
            const unsigned target = (unsigned)(t + 1) * NBLK;
            while (__hip_atomic_load(bar_cnt, __ATOMIC_ACQUIRE,
                                     __HIP_MEMORY_SCOPE_AGENT) < target) {
                __builtin_amdgcn_s_sleep(2);
            }
        }
        __syncthreads();
        __threadfence();          // invalidate so next staging sees fresh h
    }
}

extern "C" void kernel_launch(void* const* d_in, const int* in_sizes, int n_in,
                              void* d_out, int out_size, void* d_ws, size_t ws_size,
                              hipStream_t stream) {
    const int*   ids = (const int*)d_in[0];    // input_ids [4][2048] int32
    const float* W   = (const float*)d_in[1];  // [1024][1024] f32
    const float* emb = (const float*)d_in[2];  // [50257][1024] f32
    float* out = (float*)d_out;                // [4][2048][1024] f32

    unsigned char* ws = (unsigned char*)d_ws;
    unsigned* bar_cnt = (unsigned*)ws;                                   // 1 KB region
    float* hbuf0 = (float*)(ws + 1024);                                  // 16 KB
    float* hbuf1 = (float*)(ws + 1024 + BATCHSZ * DM * sizeof(float));   // 16 KB

    // deterministic per-launch init: barrier counter = 0, h[-1] = 0
    hipMemsetAsync(ws, 0, 1024 + BATCHSZ * DM * sizeof(float), stream);

    hipFuncSetAttribute((const void*)recur_wmma_f32_kernel,
                        hipFuncAttributeMaxDynamicSharedMemorySize, LDS_TOTAL_B);

    recur_wmma_f32_kernel<<<dim3(NBLK), dim3(NTHREADS), LDS_TOTAL_B, stream>>>(
        ids, W, emb, out, bar_cnt, hbuf0, hbuf1);
}